// MHSA_fl_bs_34359738368593
// MI455X (gfx1250) — compile-verified
//
#include <hip/hip_runtime.h>
#include <hip/hip_bf16.h>

// Collapsed math: softmax over the (identical) head axis == constant 1/16,
// so out[k,b,:] = (1/16)*(sum_q vp[b,q,:]) @ Wb.T + bb, broadcast over Sk.
// Only v/Wv/bv/Wb/bb are ever read; q/k/Wq/Wk do not affect the output.

#define BATCH  4
#define SQ     1024
#define DMODEL 1024
#define DH     64

typedef __attribute__((ext_vector_type(2))) float v2f;
typedef __attribute__((ext_vector_type(8))) float v8f;
typedef __attribute__((address_space(3))) float lds_f;

// ---- 1) zero the vs accumulator (4*1024 floats) --------------------------
__global__ void k_zero_vs(float* __restrict__ vs) {
    vs[blockIdx.x * 256 + threadIdx.x] = 0.0f;
}

// ---- 2) vs[b,d] = sum_q v[b,q,d]  (reads 16 MB, coalesced over d) --------
__global__ void k_reduce_v(const float* __restrict__ v, float* __restrict__ vs) {
    const int d  = blockIdx.x * 256 + threadIdx.x;   // grid.x = 4  -> D
    const int b  = blockIdx.y;                       // grid.y = 4  -> B
    const int q0 = blockIdx.z * 64;                  // grid.z = 16 -> Sq chunks
    const float* p = v + ((size_t)b * SQ + q0) * DMODEL + d;
    float s = 0.0f;
#pragma unroll 8
    for (int q = 0; q < 64; ++q) s += p[(size_t)q * DMODEL];
    atomicAdd(&vs[b * DMODEL + d], s);
}

// ---- 3) S[b,j] = (vs[b,:]·Wv[j,:] + 1024*bv[j]) / 16 ---------------------
__global__ void k_proj_v(const float* __restrict__ vs, const float* __restrict__ Wv,
                         const float* __restrict__ bv, float* __restrict__ S) {
    const int t = threadIdx.x;         // 256 threads
    const int b = t >> 6;              // 0..3
    const int j = t & 63;              // 0..63
    const float4* a = (const float4*)(vs + b * DMODEL);
    const float4* w = (const float4*)(Wv + (size_t)j * DMODEL);
    float s = 0.0f;
#pragma unroll 4
    for (int i = 0; i < DMODEL / 4; ++i) {
        float4 x = a[i], y = w[i];
        s += x.x * y.x + x.y * y.y + x.z * y.z + x.w * y.w;
    }
    S[b * DH + j] = (s + 1024.0f * bv[j]) * (1.0f / 16.0f);
}

// ---- 4) r[b,n] = S[b,:] @ Wb.T[:,n] + bb[n]  via V_WMMA_F32_16X16X4_F32 --
// M=4 padded to 16 (zero rows), K=64 (16 WMMA steps), one 16-wide N-tile per
// wave; 64 waves cover N=1024. The padded A-tile lives in LDS (shared by all
// 8 waves of the block) and is filled with an async-to-LDS load.
__global__ void k_gemm_wmma(const float* __restrict__ S, const float* __restrict__ Wb,
                            const float* __restrict__ bb, float* __restrict__ r) {
    __shared__ __align__(16) float ldsA[16 * DH];    // padded A: [16][64] f32

    const int tid  = threadIdx.x;
    const int wave = (int)((blockIdx.x * blockDim.x + tid) >> 5);  // 0..63
    const int lane = tid & 31;
    const int n0   = wave * 16;        // output column base
    const int m    = lane & 15;        // A row / B-C column within tile
    const int hi   = lane >> 4;        // lane half -> K sub-pair select

    // Zero-fill the whole padded tile (256 threads x float4 = 4 KB).
    ((float4*)ldsA)[tid] = make_float4(0.f, 0.f, 0.f, 0.f);
    __syncthreads();

    // Async-copy the 4 valid rows (256 floats, contiguous) from global S into
    // LDS: 64 lanes x b128. Tracked by ASYNCcnt, drained before the barrier.
    if (tid < 64) {
        lds_f* lp = (lds_f*)ldsA + tid * 4;          // 32-bit LDS address
        const float* gp = S + tid * 4;
        asm volatile("global_load_async_to_lds_b128 %0, %1, off"
                     :: "v"(lp), "v"(gp) : "memory");
    }
    asm volatile("s_wait_asynccnt 0x0" ::: "memory");
    __syncthreads();

    // Preload all A fragments from LDS (ds_load_b64, uniform, no divergence).
    // A layout (32-bit A 16x4): VGPR0 = {K=k0 | K=k0+2}, VGPR1 = {K=k0+1 | K=k0+3}.
    v2f areg[16];
    {
        const int aoff = m * DH + hi * 2;            // element index, even
#pragma unroll
        for (int k0 = 0; k0 < 16; ++k0)
            areg[k0] = *(const v2f*)&ldsA[aoff + k0 * 4];
    }

    // Preload all B fragments: B[k][n] = Wb[(n0+n)*DH + k]; this lane streams
    // K-pairs from row (n0+m) starting at k = hi*2 (8-byte aligned).
    v2f breg[16];
    {
        const float* wb = Wb + (size_t)(n0 + m) * DH + hi * 2;
        __builtin_prefetch(wb, 0, 3);                // global_prefetch_b8
#pragma unroll
        for (int k0 = 0; k0 < 16; ++k0)
            breg[k0] = *(const v2f*)&wb[k0 * 4];
    }

    // 16 back-to-back WMMA accumulations (full EXEC, no memory stalls).
    v8f c = {};
#pragma unroll
    for (int k0 = 0; k0 < 16; ++k0)
        c = __builtin_amdgcn_wmma_f32_16x16x4_f32(
                /*neg_a=*/false, areg[k0], /*neg_b=*/false, breg[k0],
                /*c_mod=*/(short)0, c, /*reuse_a=*/false, /*reuse_b=*/false);

    // C layout: lane l<16 holds (M=vgprIdx, N=n0+l); rows 0..3 are the batch.
    if (lane < 16) {
        const float bias = bb[n0 + lane];
#pragma unroll
        for (int mm = 0; mm < BATCH; ++mm)
            r[mm * DMODEL + n0 + lane] = c[mm] + bias;
    }
}

// ---- 5) out[k,b,d] = r[b,d]  (writes 16 MB, float4, fully coalesced) -----
__global__ void k_broadcast(const float4* __restrict__ r4, float4* __restrict__ out4) {
    const size_t i = (size_t)blockIdx.x * 256 + threadIdx.x; // 1M float4 stores
    out4[i] = r4[i & 1023];                                  // 1024 float4 per Sk row
}

extern "C" void kernel_launch(void* const* d_in, const int* in_sizes, int n_in,
                              void* d_out, int out_size, void* d_ws, size_t ws_size,
                              hipStream_t stream) {
    (void)in_sizes; (void)n_in; (void)out_size; (void)ws_size;
    // setup_inputs order: q(0) k(1) h(2) w(3) v(4) Wq(5) bq(6) Wk(7) bk(8)
    //                     Wv(9) bv(10) Wb(11) bb(12)
    const float* v  = (const float*)d_in[4];
    const float* Wv = (const float*)d_in[9];
    const float* bv = (const float*)d_in[10];
    const float* Wb = (const float*)d_in[11];
    const float* bb = (const float*)d_in[12];

    float* ws = (float*)d_ws;
    float* vs = ws;                 // [4,1024]
    float* S  = ws + 4096;          // [4,64]   (pre-scaled by 1/16)
    float* r  = ws + 4096 + 256;    // [4,1024] (16-byte aligned: 17408 B)

    k_zero_vs  <<<16, 256, 0, stream>>>(vs);
    k_reduce_v <<<dim3(4, 4, 16), 256, 0, stream>>>(v, vs);
    k_proj_v   <<<1, 256, 0, stream>>>(vs, Wv, bv, S);
    k_gemm_wmma<<<8, 256, 0, stream>>>(S, Wb, bb, r);
    k_broadcast<<<4096, 256, 0, stream>>>((const float4*)r, (float4*)d_out);
}